// BasicModel_71330816852203
// MI455X (gfx1250) — compile-verified
//
#include <hip/hip_runtime.h>
#include <math.h>

typedef float v2f __attribute__((ext_vector_type(2)));
typedef float v8f __attribute__((ext_vector_type(8)));

#define Bv   16
#define Nv   64
#define NFv  64
#define EFv  32
#define Mv   64
#define ADv  4
#define OUTv 16

// ---- CDNA5 f32 WMMA: D = A(16x4) x B(4x16) + C(16x16) ----
__device__ __forceinline__ v8f wmma_acc(v8f c, v2f a, v2f b) {
    return __builtin_amdgcn_wmma_f32_16x16x4_f32(
        /*neg_a=*/false, a, /*neg_b=*/false, b,
        /*c_mod=*/(short)0, c, /*reuse_a=*/false, /*reuse_b=*/false);
}

// T[b,j,e,m] = sum_n afm[b,j,n] * W_edge[e, m*64+n]
// one wave per (e, row-tile of bj, col-tile of m); K = NF = 64
__global__ void k_edge_T(const float* __restrict__ afm,
                         const float* __restrict__ W_edge,
                         float* __restrict__ T) {
    const int wid  = blockIdx.x * (blockDim.x >> 5) + (threadIdx.x >> 5); // [0,8192)
    const int lane = threadIdx.x & 31;
    const int row  = lane & 15;        // M-row / N-col within tile
    const int lh   = lane >> 4;        // half-wave (selects K pair)
    const int e  = wid >> 8;           // [0,32)
    const int rt = (wid >> 2) & 63;    // bj tile [0,64)
    const int ct = wid & 3;            // m tile  [0,4)

    const float* Ap = afm + (rt * 16 + row) * NFv;                 // A(row,n), stride 1 in n
    const float* Bp = W_edge + e * (Mv * NFv) + (ct * 16 + row) * NFv; // B(n,col)=W_edge[e][col*64+n]

    v8f c = {};
    for (int k0 = 0; k0 < NFv; k0 += 4) {
        const int k = k0 + 2 * lh;
        v2f a = { Ap[k], Ap[k + 1] };
        v2f b = { Bp[k], Bp[k + 1] };
        c = wmma_acc(c, a, b);
    }
    float* Op = T + (rt * 16 + 8 * lh) * (EFv * Mv) + e * Mv + ct * 16 + row;
#pragma unroll
    for (int r = 0; r < 8; ++r) Op[r * (EFv * Mv)] = c[r];
}

// msgbias[bj,m] = sum_n afm[bj,n] * b_edge[m*64+n]
__global__ void k_msgbias(const float* __restrict__ afm,
                          const float* __restrict__ b_edge,
                          float* __restrict__ msgb) {
    const int tid = blockIdx.x * blockDim.x + threadIdx.x; // 65536
    const int bj = tid >> 6, m = tid & 63;
    float s = 0.f;
    for (int n = 0; n < NFv; ++n) s += afm[bj * NFv + n] * b_edge[m * NFv + n];
    msgb[tid] = s;
}

// msg[b,i,j,m] = sum_e bfm[b,i,j,e] * T[b,j,e,m] + msgb[b,j,m]
// one wave per (bj, i-tile, m-tile); K = EF = 32
__global__ void k_msg(const float* __restrict__ bfm,
                      const float* __restrict__ T,
                      const float* __restrict__ msgb,
                      float* __restrict__ msg) {
    const int wid  = blockIdx.x * (blockDim.x >> 5) + (threadIdx.x >> 5); // [0,16384)
    const int lane = threadIdx.x & 31;
    const int row  = lane & 15;
    const int lh   = lane >> 4;
    const int bj = wid >> 4;           // [0,1024)
    const int t  = wid & 15;
    const int it = t >> 2, mt = t & 3;
    const int b = bj >> 6, j = bj & 63;

    const float* Ap = bfm + ((size_t)((b * Nv + it * 16 + row) * Nv + j)) * EFv; // A(i',e)
    const float* Bp = T + bj * (EFv * Mv) + mt * 16 + row;                       // B(e,m), k-stride 64

    v8f c = {};
    for (int k0 = 0; k0 < EFv; k0 += 4) {
        const int k = k0 + 2 * lh;
        v2f a = { Ap[k], Ap[k + 1] };
        v2f b2 = { Bp[k * Mv], Bp[(k + 1) * Mv] };
        c = wmma_acc(c, a, b2);
    }
    const float bias = msgb[bj * Mv + mt * 16 + row];
    float* Op = msg + ((size_t)((b * Nv + it * 16 + 8 * lh) * Nv + j)) * Mv + mt * 16 + row;
#pragma unroll
    for (int r = 0; r < 8; ++r) Op[(size_t)r * (Nv * Mv)] = c[r] + bias;
}

// agg[bi, k*64+m] = sum_j adj[bi, j, k] * msg[bi, j, m]   (only 4 rows -> VALU)
__global__ void k_agg(const float* __restrict__ adj,
                      const float* __restrict__ msg,
                      float* __restrict__ agg) {
    const int bi = blockIdx.x;           // [0,1024)
    const int t  = threadIdx.x;          // 256
    const int k = t >> 6, m = t & 63;
    const float* ap = adj + bi * (Nv * ADv) + k;
    const float* mp = msg + (size_t)bi * (Nv * Mv) + m;
    float s = 0.f;
    for (int j = 0; j < Nv; ++j) s += ap[j * ADv] * mp[j * Mv];
    agg[bi * (ADv * Mv) + t] = s;
}

// W_comb[km,f] = sum_m W_agg[km,m]*W_x[m,f];  biasc[f] = b_x[f] + sum_m b_agg[m]*W_x[m,f]
__global__ void k_wcomb(const float* __restrict__ W_agg, const float* __restrict__ W_x,
                        const float* __restrict__ b_agg, const float* __restrict__ b_x,
                        float* __restrict__ Wc, float* __restrict__ biasc) {
    const int tid = blockIdx.x * blockDim.x + threadIdx.x;
    const int NW = (ADv * Mv) * (3 * NFv); // 256*192
    if (tid < NW) {
        const int km = tid / 192, f = tid % 192;
        float s = 0.f;
        for (int m = 0; m < Mv; ++m) s += W_agg[km * Mv + m] * W_x[m * 192 + f];
        Wc[tid] = s;
    } else if (tid < NW + 192) {
        const int f = tid - NW;
        float s = b_x[f];
        for (int m = 0; m < Mv; ++m) s += b_agg[m] * W_x[m * 192 + f];
        biasc[f] = s;
    }
}

// Out[1024 x Ncols] = A[1024 x K] * Bm[K x Ncols] + bias[Ncols]; A packed row-major (lda=K)
__global__ void k_gemm_bias(const float* __restrict__ A, const float* __restrict__ Bm,
                            const float* __restrict__ bias, float* __restrict__ Out,
                            int K, int Ncols, int ntN) {
    const int wid  = blockIdx.x * (blockDim.x >> 5) + (threadIdx.x >> 5);
    if (wid >= 64 * ntN) return;       // wave-uniform guard (EXEC stays all-ones)
    const int lane = threadIdx.x & 31;
    const int row  = lane & 15;
    const int lh   = lane >> 4;
    const int rt = wid / ntN, ct = wid % ntN;

    const float* Ap = A + (rt * 16 + row) * K;
    const float* Bp = Bm + ct * 16 + row;
    v8f c = {};
    for (int k0 = 0; k0 < K; k0 += 4) {
        const int k = k0 + 2 * lh;
        v2f a  = { Ap[k], Ap[k + 1] };
        v2f b2 = { Bp[k * Ncols], Bp[(k + 1) * Ncols] };
        c = wmma_acc(c, a, b2);
    }
    const float bv = bias[ct * 16 + row];
    float* Op = Out + (rt * 16 + 8 * lh) * Ncols + ct * 16 + row;
#pragma unroll
    for (int r = 0; r < 8; ++r) Op[r * Ncols] = c[r] + bv;
}

__device__ __forceinline__ float sigm(float x) { return 1.0f / (1.0f + expf(-x)); }

// GRU gate update: h_out = ((1-z)*ng + z*h_in) * mask
__global__ void k_gru(const float* __restrict__ gi, const float* __restrict__ gh,
                      const float* __restrict__ h_in, const float* __restrict__ mask,
                      float* __restrict__ h_out) {
    const int tid = blockIdx.x * blockDim.x + threadIdx.x; // 65536
    const int bi = tid >> 6, n = tid & 63;
    const int o = bi * 192 + n;
    const float r  = sigm(gi[o] + gh[o]);
    const float z  = sigm(gi[o + 64] + gh[o + 64]);
    const float ng = tanhf(gi[o + 128] + r * gh[o + 128]);
    h_out[tid] = ((1.0f - z) * ng + z * h_in[tid]) * mask[bi];
}

// out[b,o] = sum_i mask * sigmoid(x@W_i+b_i) * (x@W_j+b_j),  x = [h | afm]
__global__ void k_readout(const float* __restrict__ h, const float* __restrict__ afm,
                          const float* __restrict__ mask,
                          const float* __restrict__ W_i, const float* __restrict__ b_i,
                          const float* __restrict__ W_j, const float* __restrict__ b_j,
                          float* __restrict__ out) {
    __shared__ float sm[256];
    const int b = blockIdx.x;          // [0,16)
    const int t = threadIdx.x;         // 256
    const int o = t >> 4, ii = t & 15;
    float acc = 0.f;
    for (int s = 0; s < 4; ++s) {
        const int i = ii + 16 * s, bi = b * Nv + i;
        float di = b_i[o], dj = b_j[o];
        for (int n = 0; n < NFv; ++n) {
            const float xv = h[bi * NFv + n];
            di += xv * W_i[n * OUTv + o];
            dj += xv * W_j[n * OUTv + o];
        }
        for (int n = 0; n < NFv; ++n) {
            const float xv = afm[bi * NFv + n];
            di += xv * W_i[(NFv + n) * OUTv + o];
            dj += xv * W_j[(NFv + n) * OUTv + o];
        }
        acc += mask[bi] * sigm(di) * dj;
    }
    sm[t] = acc;
    __syncthreads();
    for (int st = 8; st > 0; st >>= 1) {
        if (ii < st) sm[t] += sm[t + st];
        __syncthreads();
    }
    if (ii == 0) out[b * OUTv + o] = sm[o * 16];
}

extern "C" void kernel_launch(void* const* d_in, const int* in_sizes, int n_in,
                              void* d_out, int out_size, void* d_ws, size_t ws_size,
                              hipStream_t stream) {
    const float* afm    = (const float*)d_in[0];
    const float* bfm    = (const float*)d_in[1];
    const float* adj    = (const float*)d_in[2];
    const float* mask   = (const float*)d_in[3];
    const float* W_edge = (const float*)d_in[4];
    const float* b_edge = (const float*)d_in[5];
    const float* W_agg  = (const float*)d_in[6];
    const float* b_agg  = (const float*)d_in[7];
    const float* W_x    = (const float*)d_in[8];
    const float* W_h    = (const float*)d_in[9];
    const float* b_x    = (const float*)d_in[10];
    const float* b_h    = (const float*)d_in[11];
    const float* W_i    = (const float*)d_in[12];
    const float* b_i    = (const float*)d_in[13];
    const float* W_j    = (const float*)d_in[14];
    const float* b_j    = (const float*)d_in[15];
    float* out = (float*)d_out;

    // workspace layout (floats); total ~7.2M floats (~29 MB)
    float* ws    = (float*)d_ws;
    float* T     = ws;                  // 16*64*32*64      = 2,097,152
    float* msgb  = T     + 2097152;     // 16*64*64         =    65,536
    float* msg   = msgb  + 65536;       // 16*64*64*64      = 4,194,304
    float* agg   = msg   + 4194304;     // 16*64*4*64       =   262,144
    float* Wc    = agg   + 262144;      // 256*192          =    49,152
    float* biasc = Wc    + 49152;       // 192 (pad 256)
    float* gi    = biasc + 256;         // 1024*192         =   196,608
    float* gh    = gi    + 196608;      // 1024*192         =   196,608
    float* h0    = gh    + 196608;      // 1024*64          =    65,536
    float* h1    = h0    + 65536;       // 1024*64          =    65,536

    k_edge_T <<<1024, 256, 0, stream>>>(afm, W_edge, T);        // 8192 waves
    k_msgbias<<< 256, 256, 0, stream>>>(afm, b_edge, msgb);
    k_msg    <<<2048, 256, 0, stream>>>(bfm, T, msgb, msg);     // 16384 waves
    k_agg    <<<1024, 256, 0, stream>>>(adj, msg, agg);
    k_wcomb  <<< 193, 256, 0, stream>>>(W_agg, W_x, b_agg, b_x, Wc, biasc);
    k_gemm_bias<<<96, 256, 0, stream>>>(agg, Wc, biasc, gi, 256, 192, 12);

    const float* hin = afm;
    float* hA = h0; float* hB = h1;
    for (int s = 0; s < 3; ++s) {
        k_gemm_bias<<<96, 256, 0, stream>>>(hin, W_h, b_h, gh, 64, 192, 12);
        k_gru<<<256, 256, 0, stream>>>(gi, gh, hin, mask, hA);
        hin = hA; float* tmp = hA; hA = hB; hB = tmp;
    }
    k_readout<<<16, 256, 0, stream>>>(hin, afm, mask, W_i, b_i, W_j, b_j, out);

    (void)in_sizes; (void)n_in; (void)out_size; (void)ws_size;
}